// DetectModel_56126632624642
// MI455X (gfx1250) — compile-verified
//
#include <hip/hip_runtime.h>
#include <hip/hip_bf16.h>

// Problem constants (from reference)
#define NTOT 50000
#define EDG  500000
#define BG   25
#define NPG  2000
#define DIM  16
#define REL  114
#define LAY  3
#define HID  16
#define MAXTILES (EDG/16 + REL)   // 31364 upper bound on 16-edge tiles
#define NEGF (-1e30f)

typedef __attribute__((ext_vector_type(2))) float v2f;
typedef __attribute__((ext_vector_type(8))) float v8f;

__device__ inline void atomicMaxFloat(float* addr, float v) {
  // valid when buffer initialized to a negative float (-1e30)
  if (v >= 0.0f) atomicMax((int*)addr, __float_as_int(v));
  else           atomicMin((unsigned int*)addr, __float_as_uint(v));
}

// ---------------- setup kernels ----------------

__global__ void k_embed(const int* __restrict__ attr, const float* __restrict__ emb,
                        float* __restrict__ x) {
  int i = blockIdx.x * blockDim.x + threadIdx.x;
  if (i >= NTOT) return;
  int a = attr[i];
  const float4* e4 = (const float4*)(emb + (size_t)a * DIM);
  float4* x4 = (float4*)(x + (size_t)i * DIM);
  x4[0] = e4[0]; x4[1] = e4[1]; x4[2] = e4[2]; x4[3] = e4[3];
}

__global__ void k_initmask(unsigned char* __restrict__ em, unsigned char* __restrict__ nm) {
  int e = blockIdx.x * blockDim.x + threadIdx.x;
  if (e >= EDG) return;
  em[e] = 1;
  if (e < NTOT) nm[e] = 1;
}

__global__ void k_hist(const int* __restrict__ et, int* __restrict__ segCnt) {
  int e = blockIdx.x * blockDim.x + threadIdx.x;
  if (e >= EDG) return;
  atomicAdd(&segCnt[et[e]], 1);
}

__global__ void k_scan(const int* __restrict__ segCnt, int* __restrict__ segStart,
                       int* __restrict__ cursor, int* __restrict__ tileOff) {
  int s = 0, t = 0;
  for (int r = 0; r < REL; ++r) {
    segStart[r] = s; cursor[r] = s; tileOff[r] = t;
    t += (segCnt[r] + 15) >> 4;
    s += segCnt[r];
  }
  segStart[REL] = s; tileOff[REL] = t;
}

__global__ void k_scatter(const int* __restrict__ et, int* __restrict__ cursor,
                          int* __restrict__ perm) {
  int e = blockIdx.x * blockDim.x + threadIdx.x;
  if (e >= EDG) return;
  int r = et[e];
  int pos = atomicAdd(&cursor[r], 1);
  perm[pos] = e;
}

__global__ void k_tilefill(const int* __restrict__ segStart, const int* __restrict__ tileOff,
                           int4* __restrict__ tiles) {
  int t = blockIdx.x * blockDim.x + threadIdx.x;
  if (t >= MAXTILES) return;
  if (t >= tileOff[REL]) return;
  int r = 0;
  while (!(t >= tileOff[r] && t < tileOff[r + 1])) r++;
  int lt = t - tileOff[r];
  int st = segStart[r] + lt * 16;
  int cnt = segStart[r + 1] - st;
  if (cnt > 16) cnt = 16;
  tiles[t] = make_int4(r, st, cnt, 0);
}

// ---------------- per-layer kernels ----------------

// Wq_r = W_r @ q, Wk_r = W_r @ k   (so logit = x_dst.Wq + x_src.Wk)
__global__ void k_wqk(const float* __restrict__ W, const float* __restrict__ q,
                      const float* __restrict__ kk, float* __restrict__ wq,
                      float* __restrict__ wk) {
  int idx = blockIdx.x * blockDim.x + threadIdx.x;
  if (idx >= REL * DIM) return;
  int r = idx >> 4, d = idx & 15;
  const float* Wr = W + (size_t)r * 256 + d * 16;
  float sq = 0.f, sk = 0.f;
  for (int f = 0; f < 16; ++f) { sq += Wr[f] * q[f]; sk += Wr[f] * kk[f]; }
  wq[idx] = sq; wk[idx] = sk;
}

__global__ void k_initnode(float* __restrict__ m, float* __restrict__ den,
                           float* __restrict__ agg) {
  int i = blockIdx.x * blockDim.x + threadIdx.x;
  if (i >= NTOT) return;
  m[i] = NEGF; den[i] = 0.f;
  float4 z = make_float4(0.f, 0.f, 0.f, 0.f);
  float4* a4 = (float4*)(agg + (size_t)i * DIM);
  a4[0] = z; a4[1] = z; a4[2] = z; a4[3] = z;
}

__global__ void k_logit(const int* __restrict__ src, const int* __restrict__ dst,
                        const int* __restrict__ et, const unsigned char* __restrict__ em,
                        const float* __restrict__ x, const float* __restrict__ wq,
                        const float* __restrict__ wk, float* __restrict__ lg,
                        float* __restrict__ m) {
  int e = blockIdx.x * blockDim.x + threadIdx.x;
  if (e >= EDG) return;
  int s = src[e], dd = dst[e], r = et[e];
  const float* xs = x + (size_t)s * DIM;
  const float* xd = x + (size_t)dd * DIM;
  const float* q = wq + r * DIM;
  const float* k2 = wk + r * DIM;
  float li = 0.f, lj = 0.f;
  for (int d = 0; d < 16; ++d) { li += xd[d] * q[d]; lj += xs[d] * k2[d]; }
  float v = li + lj;
  v = v > 0.f ? v : 0.2f * v;            // leaky_relu 0.2
  v = em[e] ? v : NEGF;
  lg[e] = v;
  atomicMaxFloat(&m[dd], v);
}

__global__ void k_alpha(const int* __restrict__ dst, const unsigned char* __restrict__ em,
                        const float* __restrict__ lg, const float* __restrict__ m,
                        float* __restrict__ aE, float* __restrict__ den) {
  int e = blockIdx.x * blockDim.x + threadIdx.x;
  if (e >= EDG) return;
  int dd = dst[e];
  float v = em[e] ? __expf(lg[e] - m[dd]) : 0.f;
  aE[e] = v;
  atomicAdd(&den[dd], v);
}

// Normalize attention weights per edge: aE[e] <- aE[e] / (den[dst[e]] + 1e-16)
// Done here (one division per EDGE) so the WMMA kernel's epilogue is mul+atomic only.
__global__ void k_norm(const int* __restrict__ dst, const float* __restrict__ den,
                       float* __restrict__ aE) {
  int e = blockIdx.x * blockDim.x + threadIdx.x;
  if (e >= EDG) return;
  aE[e] = aE[e] / (den[dst[e]] + 1e-16f);
}

// One wave per 16-edge tile (all edges share relation r):
//   Mj(16x16) = Xsrc(16x16) @ W_r(16x16) via 4x V_WMMA_F32_16X16X4_F32,
// then scatter alpha-weighted rows into agg[dst].
__global__ __launch_bounds__(256) void k_gemm(
    const int4* __restrict__ tiles, const int* __restrict__ tileOff,
    const int* __restrict__ perm, const int* __restrict__ src,
    const int* __restrict__ dst, const float* __restrict__ x,
    const float* __restrict__ W, const float* __restrict__ aE,
    float* __restrict__ agg) {
  int wave = (blockIdx.x * blockDim.x + threadIdx.x) >> 5;
  int lane = threadIdx.x & 31;
  if (wave >= tileOff[REL]) return;           // uniform per wave (EXEC stays full)
  int4 td = tiles[wave];
  int r = td.x, st = td.y, cnt = td.z;
  const float* Wr = W + (size_t)r * 256;

  int half = lane >> 4;                        // 0: lanes 0-15, 1: lanes 16-31
  int col  = lane & 15;                        // N (feature) index / M row index for A
  int mrow = col;                              // A-matrix row handled by this lane pair
  int e = (mrow < cnt) ? perm[st + mrow] : -1;
  const float* xr = x + (size_t)((e >= 0) ? src[e] : 0) * DIM;

  v2f av[4], bv[4];
#pragma unroll
  for (int kk = 0; kk < 4; ++kk) {
    int d0 = 4 * kk + 2 * half;
    float2 t2 = (e >= 0) ? *(const float2*)(xr + d0) : make_float2(0.f, 0.f);
    av[kk][0] = t2.x; av[kk][1] = t2.y;        // A[m][d0], A[m][d0+1]
    bv[kk][0] = Wr[d0 * 16 + col];             // B[d0][col]
    bv[kk][1] = Wr[(d0 + 1) * 16 + col];       // B[d0+1][col]
  }

  v8f c = {0.f, 0.f, 0.f, 0.f, 0.f, 0.f, 0.f, 0.f};
  c = __builtin_amdgcn_wmma_f32_16x16x4_f32(false, av[0], false, bv[0], (short)0, c, false, false);
  c = __builtin_amdgcn_wmma_f32_16x16x4_f32(false, av[1], false, bv[1], (short)0, c, false, false);
  c = __builtin_amdgcn_wmma_f32_16x16x4_f32(false, av[2], false, bv[2], (short)0, c, false, false);
  c = __builtin_amdgcn_wmma_f32_16x16x4_f32(false, av[3], false, bv[3], (short)0, c, false, false);

#pragma unroll
  for (int j = 0; j < 8; ++j) {
    int mr = j + 8 * half;                     // D row held by this vgpr on this lane
    if (mr < cnt) {
      int ee = perm[st + mr];
      int dd = dst[ee];
      atomicAdd(&agg[(size_t)dd * DIM + col], aE[ee] * c[j]);
    }
  }
}

__global__ void k_nodepost(const float* __restrict__ agg, const float* __restrict__ bl,
                           const unsigned char* __restrict__ nm, float* __restrict__ x) {
  int i = blockIdx.x * blockDim.x + threadIdx.x;
  if (i >= NTOT) return;
  float msk = nm[i] ? 1.f : 0.f;
  for (int d = 0; d < 16; ++d) {
    float v = agg[(size_t)i * DIM + d] + bl[d];
    v = v > 0.f ? v : 0.f;
    x[(size_t)i * DIM + d] = v * msk;
  }
}

// one block per (graph, dim): masked mean + masked max
__global__ void k_pool(const float* __restrict__ x, const unsigned char* __restrict__ nm,
                       float* __restrict__ feats, int l) {
  int b = blockIdx.x >> 4;
  int dim = blockIdx.x & 15;
  __shared__ float ssum[256];
  __shared__ float smax[256];
  __shared__ int scnt[256];
  int tid = threadIdx.x;
  float lsum = 0.f, lmax = NEGF; int lcnt = 0;
  for (int i = tid; i < NPG; i += blockDim.x) {
    int node = b * NPG + i;
    float v = x[(size_t)node * DIM + dim];     // masked nodes are already zero
    lsum += v;
    if (nm[node]) { lmax = fmaxf(lmax, v); lcnt++; }
  }
  ssum[tid] = lsum; smax[tid] = lmax; scnt[tid] = lcnt;
  __syncthreads();
  for (int s = 128; s > 0; s >>= 1) {
    if (tid < s) {
      ssum[tid] += ssum[tid + s];
      smax[tid] = fmaxf(smax[tid], smax[tid + s]);
      scnt[tid] += scnt[tid + s];
    }
    __syncthreads();
  }
  if (tid == 0) {
    float cnt = fmaxf((float)scnt[0], 1.f);
    feats[b * 96 + 32 * l + dim] = ssum[0] / cnt;
    feats[b * 96 + 32 * l + 16 + dim] = smax[0];
  }
}

// TopK pooling: exact kth-largest via rank count in LDS; scale x by score*keep
__global__ void k_topk(float* __restrict__ x, unsigned char* __restrict__ nm,
                       const float* __restrict__ w, int kth) {
  int b = blockIdx.x;
  __shared__ float ss[NPG];
  __shared__ float wsh[16];
  __shared__ float wn;
  int tid = threadIdx.x;
  if (tid < 16) wsh[tid] = w[tid];
  __syncthreads();
  if (tid == 0) {
    float s2 = 0.f;
    for (int d = 0; d < 16; ++d) s2 += wsh[d] * wsh[d];
    wn = sqrtf(s2) + 1e-16f;
  }
  __syncthreads();
  for (int i = tid; i < NPG; i += blockDim.x) {
    int node = b * NPG + i;
    const float* xr = x + (size_t)node * DIM;
    float dv = 0.f;
    for (int d = 0; d < 16; ++d) dv += xr[d] * wsh[d];
    float sc = tanhf(dv / wn);
    ss[i] = nm[node] ? sc : NEGF;
  }
  __syncthreads();
  for (int i = tid; i < NPG; i += blockDim.x) {
    int node = b * NPG + i;
    float si = ss[i];
    int cg = 0;
    for (int j = 0; j < NPG; ++j) cg += (ss[j] > si) ? 1 : 0;
    int keep = (nm[node] && cg < kth) ? 1 : 0;   // == (s_i >= kth largest)
    nm[node] = (unsigned char)keep;
    float sc = keep ? si : 0.f;
    float* xr = x + (size_t)node * DIM;
    for (int d = 0; d < 16; ++d) xr[d] *= sc;
  }
}

__global__ void k_emask(const int* __restrict__ src, const int* __restrict__ dst,
                        const unsigned char* __restrict__ nm, unsigned char* __restrict__ em) {
  int e = blockIdx.x * blockDim.x + threadIdx.x;
  if (e >= EDG) return;
  if (em[e]) {
    if (!nm[src[e]] || !nm[dst[e]]) em[e] = 0;
  }
}

// Sequential GRU over the 25 graph features + 2-layer MLP head (single block)
__global__ void k_head(const float* __restrict__ feats, const float* __restrict__ Wih,
                       const float* __restrict__ Whh, const float* __restrict__ bih,
                       const float* __restrict__ bhh, const float* __restrict__ W1,
                       const float* __restrict__ b1, const float* __restrict__ W2,
                       const float* __restrict__ b2, float* __restrict__ out) {
  __shared__ float h[HID];
  __shared__ float gi[48], gh[48];
  __shared__ float hs[BG * HID];
  int t = threadIdx.x;
  if (t < HID) h[t] = 0.f;
  __syncthreads();
  for (int b = 0; b < BG; ++b) {
    const float* g = feats + b * 96;
    if (t < 48) {
      float s = bih[t];
      for (int j = 0; j < 96; ++j) s += Wih[t * 96 + j] * g[j];
      gi[t] = s;
      float s2 = bhh[t];
      for (int j = 0; j < 16; ++j) s2 += Whh[t * 16 + j] * h[j];
      gh[t] = s2;
    }
    __syncthreads();
    float hnew = 0.f;
    if (t < HID) {
      float rr = 1.f / (1.f + __expf(-(gi[t] + gh[t])));
      float zz = 1.f / (1.f + __expf(-(gi[16 + t] + gh[16 + t])));
      float nn = tanhf(gi[32 + t] + rr * gh[32 + t]);
      hnew = (1.f - zz) * nn + zz * h[t];
    }
    __syncthreads();
    if (t < HID) { h[t] = hnew; hs[b * HID + t] = hnew; }
    __syncthreads();
  }
  if (t < BG) {
    float o1[4];
    for (int j = 0; j < 4; ++j) {
      float s = b1[j];
      for (int d = 0; d < 16; ++d) s += W1[j * 16 + d] * hs[t * HID + d];
      o1[j] = s > 0.f ? s : 0.f;
    }
    float s = b2[0];
    for (int j = 0; j < 4; ++j) s += W2[j] * o1[j];
    out[t] = s > 0.f ? s : 0.f;
  }
}

// ---------------- launch ----------------

extern "C" void kernel_launch(void* const* d_in, const int* in_sizes, int n_in,
                              void* d_out, int out_size, void* d_ws, size_t ws_size,
                              hipStream_t stream) {
  (void)in_sizes; (void)n_in; (void)out_size; (void)ws_size;
  const int*   node_attr = (const int*)d_in[0];
  const int*   eidx      = (const int*)d_in[1];
  const int*   src       = eidx;
  const int*   dstp      = eidx + EDG;
  const int*   et        = (const int*)d_in[2];
  const float* emb       = (const float*)d_in[3];
  const float* gnnW      = (const float*)d_in[4];
  const float* gnnq      = (const float*)d_in[5];
  const float* gnnk      = (const float*)d_in[6];
  const float* gnnb      = (const float*)d_in[7];
  const float* poolw     = (const float*)d_in[8];
  const float* Wih       = (const float*)d_in[9];
  const float* Whh       = (const float*)d_in[10];
  const float* bih       = (const float*)d_in[11];
  const float* bhh       = (const float*)d_in[12];
  const float* W1        = (const float*)d_in[13];
  const float* b1        = (const float*)d_in[14];
  const float* W2        = (const float*)d_in[15];
  const float* b2        = (const float*)d_in[16];

  char* p = (char*)d_ws;
  auto alloc = [&](size_t bytes) -> char* {
    char* r = p;
    p += (bytes + 255) & ~(size_t)255;
    return r;
  };
  float*         x        = (float*)alloc((size_t)NTOT * DIM * 4);
  float*         agg      = (float*)alloc((size_t)NTOT * DIM * 4);
  float*         mbuf     = (float*)alloc((size_t)NTOT * 4);
  float*         den      = (float*)alloc((size_t)NTOT * 4);
  float*         lg       = (float*)alloc((size_t)EDG * 4);
  float*         aE       = (float*)alloc((size_t)EDG * 4);
  int*           perm     = (int*)alloc((size_t)EDG * 4);
  unsigned char* em       = (unsigned char*)alloc(EDG);
  unsigned char* nm       = (unsigned char*)alloc(NTOT);
  int*           segCnt   = (int*)alloc(REL * 4);
  int*           segStart = (int*)alloc((REL + 1) * 4);
  int*           cursor   = (int*)alloc(REL * 4);
  int*           tileOff  = (int*)alloc((REL + 1) * 4);
  int4*          tiles    = (int4*)alloc((size_t)MAXTILES * 16);
  float*         wq       = (float*)alloc(REL * DIM * 4);
  float*         wk       = (float*)alloc(REL * DIM * 4);
  float*         feats    = (float*)alloc(BG * 96 * 4);

  const int TB = 256;
  hipMemsetAsync(segCnt, 0, REL * 4, stream);
  k_embed<<<(NTOT + TB - 1) / TB, TB, 0, stream>>>(node_attr, emb, x);
  k_initmask<<<(EDG + TB - 1) / TB, TB, 0, stream>>>(em, nm);
  k_hist<<<(EDG + TB - 1) / TB, TB, 0, stream>>>(et, segCnt);
  k_scan<<<1, 1, 0, stream>>>(segCnt, segStart, cursor, tileOff);
  k_scatter<<<(EDG + TB - 1) / TB, TB, 0, stream>>>(et, cursor, perm);
  k_tilefill<<<(MAXTILES + TB - 1) / TB, TB, 0, stream>>>(segStart, tileOff, tiles);

  for (int l = 0; l < LAY; ++l) {
    const float* Wl = gnnW + (size_t)l * REL * DIM * DIM;
    k_wqk<<<(REL * DIM + TB - 1) / TB, TB, 0, stream>>>(Wl, gnnq + l * DIM, gnnk + l * DIM, wq, wk);
    k_initnode<<<(NTOT + TB - 1) / TB, TB, 0, stream>>>(mbuf, den, agg);
    k_logit<<<(EDG + TB - 1) / TB, TB, 0, stream>>>(src, dstp, et, em, x, wq, wk, lg, mbuf);
    k_alpha<<<(EDG + TB - 1) / TB, TB, 0, stream>>>(dstp, em, lg, mbuf, aE, den);
    k_norm<<<(EDG + TB - 1) / TB, TB, 0, stream>>>(dstp, den, aE);
    k_gemm<<<((size_t)MAXTILES * 32 + TB - 1) / TB, TB, 0, stream>>>(
        tiles, tileOff, perm, src, dstp, x, Wl, aE, agg);
    k_nodepost<<<(NTOT + TB - 1) / TB, TB, 0, stream>>>(agg, gnnb + l * DIM, nm, x);
    k_pool<<<BG * DIM, TB, 0, stream>>>(x, nm, feats, l);
    if (l < LAY - 1) {
      int kth = (l == 0) ? 1600 : 1280;   // ceil(0.8*2000), ceil(0.8*1600)
      k_topk<<<BG, 1024, 0, stream>>>(x, nm, poolw + l * DIM, kth);
      k_emask<<<(EDG + TB - 1) / TB, TB, 0, stream>>>(src, dstp, nm, em);
    }
  }
  k_head<<<1, 64, 0, stream>>>(feats, Wih, Whh, bih, bhh, W1, b1, W2, b2, (float*)d_out);
}